// NGFLoss_40200893891260
// MI455X (gfx1250) — compile-verified
//
#include <hip/hip_runtime.h>
#include <hip/hip_bf16.h>
#include <stdint.h>

#define WW 512
#define HH 512

typedef __attribute__((ext_vector_type(2))) float v2f;
typedef __attribute__((ext_vector_type(8))) float v8f;

__device__ __forceinline__ float wave_sum32(float a) {
  #pragma unroll
  for (int m = 16; m; m >>= 1) a += __shfl_xor(a, m, 32);
  return a;
}

// One block per (b, h) row. 128 threads, 4 pixels (float4) per thread.
__global__ __launch_bounds__(128) void ngf_row_kernel(
    const float* __restrict__ I0, const float* __restrict__ I1,
    float* __restrict__ partial)
{
  __shared__ __align__(16) float s0[WW];
  __shared__ __align__(16) float s1[WW];
  __shared__ float bsum[4];

  const int row = blockIdx.x;              // b*HH + h
  const int h   = row & (HH - 1);
  const int t   = threadIdx.x;             // 0..127
  const int w4  = t << 2;                  // 0..508 step 4

  const float* r0c = I0 + (size_t)row * WW;
  const float* r1c = I1 + (size_t)row * WW;
  const int dm = (h == 0)      ? 0 : -WW;  // one-sided diff at edges folds
  const int dp = (h == HH - 1) ? 0 :  WW;  // into clamp(h+/-1)

  // Stage the two center rows into LDS via gfx1250 async-to-LDS DMA.
  {
    unsigned l0 = (unsigned)(uintptr_t)(&s0[w4]);   // flat->LDS: low 32 bits
    unsigned l1 = (unsigned)(uintptr_t)(&s1[w4]);
    const float* g0 = r0c + w4;
    const float* g1 = r1c + w4;
    asm volatile("global_load_async_to_lds_b128 %0, %1, off"
                 :: "v"(l0), "v"(g0) : "memory");
    asm volatile("global_load_async_to_lds_b128 %0, %1, off"
                 :: "v"(l1), "v"(g1) : "memory");
  }

  // Neighbor rows (no lateral offset needed) straight to VGPRs, b128 loads.
  const float4 a0m = *(const float4*)(r0c + dm + w4);
  const float4 a0p = *(const float4*)(r0c + dp + w4);
  const float4 a1m = *(const float4*)(r1c + dm + w4);
  const float4 a1p = *(const float4*)(r1c + dp + w4);

  asm volatile("s_wait_asynccnt 0" ::: "memory");
  __syncthreads();

  const float gx0_[4] = {a0p.x - a0m.x, a0p.y - a0m.y, a0p.z - a0m.z, a0p.w - a0m.w};
  const float gx1_[4] = {a1p.x - a1m.x, a1p.y - a1m.y, a1p.z - a1m.z, a1p.w - a1m.w};

  float acc = 0.0f;
  #pragma unroll
  for (int j = 0; j < 4; ++j) {
    const int w  = w4 + j;
    const int wl = (w == 0)      ? 0      : w - 1;
    const int wr = (w == WW - 1) ? WW - 1 : w + 1;
    const float gy0 = s0[wr] - s0[wl];
    const float gy1 = s1[wr] - s1[wl];
    const float gx0 = gx0_[j], gx1 = gx1_[j];
    // dot^2 of unit gradients == num^2 / (d0*d1); no sqrt needed.
    const float num = gx0 * gx1 + gy0 * gy1;
    const float d0  = gx0 * gx0 + gy0 * gy0 + 1e-10f;
    const float d1  = gx1 * gx1 + gy1 * gy1 + 1e-10f;
    acc += (num * num) * __builtin_amdgcn_rcpf(d0 * d1);
  }

  acc = wave_sum32(acc);
  if ((t & 31) == 0) bsum[t >> 5] = acc;
  __syncthreads();
  if (t == 0) partial[row] = (bsum[0] + bsum[1]) + (bsum[2] + bsum[3]);
}

// Deterministic fixed-order reduction of the 32768 row partials.
__global__ __launch_bounds__(1024) void ngf_reduce_kernel(
    const float* __restrict__ partial, int n, float* __restrict__ out)
{
  __shared__ float wsum[32];
  const int t = threadIdx.x;

  float a = 0.0f;
  for (int i = t; i < n; i += 1024) a += partial[i];
  a = wave_sum32(a);
  if ((t & 31) == 0) wsum[t >> 5] = a;
  __syncthreads();

  if (t < 32) {
    // Fold the 32 wave sums with one f32 WMMA: A = ones (16x4), B holds the
    // 32 values (4x16, padded with zeros) -> D[m][n] = colsum[n] for every m.
    // Each element contributes exactly once; D rows replicate, so summing
    // c[0] over all 32 lanes yields 2x the grand total. Exact f32 FMA chain,
    // fixed order -> deterministic.
    v2f A; A[0] = 1.0f; A[1] = 1.0f;
    v2f B; B[0] = wsum[t]; B[1] = 0.0f;
    v8f C = {};
    C = __builtin_amdgcn_wmma_f32_16x16x4_f32(
        /*neg_a=*/false, A, /*neg_b=*/false, B,
        /*c_mod=*/(short)0, C, /*reuse_a=*/false, /*reuse_b=*/false);
    float v = wave_sum32(C[0]) * 0.5f;
    if (t == 0) {
      const float invN = 1.0f / (64.0f * 512.0f * 512.0f);  // exact 2^-24
      out[0] = 1.0f - v * invN;
    }
  }
}

extern "C" void kernel_launch(void* const* d_in, const int* in_sizes, int n_in,
                              void* d_out, int out_size, void* d_ws, size_t ws_size,
                              hipStream_t stream) {
  const float* I0 = (const float*)d_in[0];
  const float* I1 = (const float*)d_in[1];
  float* out      = (float*)d_out;
  float* partial  = (float*)d_ws;          // 32768 floats = 128 KB scratch

  const int nrows = 64 * HH;               // 32768 (b,h) rows
  ngf_row_kernel<<<nrows, 128, 0, stream>>>(I0, I1, partial);
  ngf_reduce_kernel<<<1, 1024, 0, stream>>>(partial, nrows, out);
}